// DHGCN_PAConv_53893249630268
// MI455X (gfx1250) — compile-verified
//
#include <hip/hip_runtime.h>
#include <float.h>
#include <math.h>

#define BB 8
#define NN 1024
#define KK 30
#define PP 27
#define MM 8
#define CH 64

typedef float v2f __attribute__((ext_vector_type(2)));
typedef float v8f __attribute__((ext_vector_type(8)));

// ---------------------------------------------------------------------------
// WMMA GEMM:  C[M,N] = A[M,K] * B(K,N) (+bias[n])
// transB is a template parameter (straight-line K-loop body).
// Wave tile = 16(M) x 64(N): one A fragment feeds 4 independent
// V_WMMA_F32_16X16X4_F32 accumulators per K-step.
// Interior path is software-pipelined: chunk k+4's fragments are loaded
// before chunk k's WMMAs issue, so waits can be partial (load/XDL overlap).
// Edge tiles use clamped addresses + value selects (no exec churn).
// ---------------------------------------------------------------------------
template <int TB>
__global__ void k_gemm_wmma(const float* __restrict__ Ap, int lda,
                            const float* __restrict__ Bp, int ldb,
                            const float* __restrict__ bias,
                            float* __restrict__ Cp, int ldc,
                            int Mm, int Nn, int Kk) {
  int lane = threadIdx.x & 31;     // wave32
  int l  = lane & 15;
  int hi = lane >> 4;
  int  m0 = blockIdx.x * 16;
  long n0 = (long)blockIdx.y * 64;
  int row = m0 + l;

  v8f acc0 = (v8f){0.f,0.f,0.f,0.f,0.f,0.f,0.f,0.f};
  v8f acc1 = acc0, acc2 = acc0, acc3 = acc0;

  bool interior = (m0 + 16 <= Mm) && (n0 + 64 <= Nn) && ((Kk & 3) == 0);

  if (interior) {
    const float* arow = Ap + (long)row * lda;
    long c0 = n0 + l, c1 = c0 + 16, c2 = c0 + 32, c3 = c0 + 48;

    auto loadA = [&](int k) -> v2f {
      int ka = k + 2 * hi;                 // A VGPR0 = K=ka, VGPR1 = K=ka+1
      v2f a; a.x = arow[ka]; a.y = arow[ka + 1]; return a;
    };
    auto loadB = [&](int k, long c) -> v2f {
      int ka = k + 2 * hi;
      v2f b;
      if (TB == 0) { b.x = Bp[(long)ka * ldb + c]; b.y = Bp[(long)(ka + 1) * ldb + c]; }
      else         { b.x = Bp[c * ldb + ka];       b.y = Bp[c * ldb + ka + 1]; }
      return b;
    };

    // prologue: load chunk 0
    v2f a_c  = loadA(0);
    v2f b0_c = loadB(0, c0), b1_c = loadB(0, c1);
    v2f b2_c = loadB(0, c2), b3_c = loadB(0, c3);

    int k = 0;
    for (; k + 4 < Kk; k += 4) {
      // prefetch chunk k+4 before consuming chunk k
      v2f a_n  = loadA(k + 4);
      v2f b0_n = loadB(k + 4, c0), b1_n = loadB(k + 4, c1);
      v2f b2_n = loadB(k + 4, c2), b3_n = loadB(k + 4, c3);
      acc0 = __builtin_amdgcn_wmma_f32_16x16x4_f32(false, a_c, false, b0_c, (short)0, acc0, false, false);
      acc1 = __builtin_amdgcn_wmma_f32_16x16x4_f32(false, a_c, false, b1_c, (short)0, acc1, false, false);
      acc2 = __builtin_amdgcn_wmma_f32_16x16x4_f32(false, a_c, false, b2_c, (short)0, acc2, false, false);
      acc3 = __builtin_amdgcn_wmma_f32_16x16x4_f32(false, a_c, false, b3_c, (short)0, acc3, false, false);
      a_c = a_n; b0_c = b0_n; b1_c = b1_n; b2_c = b2_n; b3_c = b3_n;
    }
    // epilogue: last chunk
    acc0 = __builtin_amdgcn_wmma_f32_16x16x4_f32(false, a_c, false, b0_c, (short)0, acc0, false, false);
    acc1 = __builtin_amdgcn_wmma_f32_16x16x4_f32(false, a_c, false, b1_c, (short)0, acc1, false, false);
    acc2 = __builtin_amdgcn_wmma_f32_16x16x4_f32(false, a_c, false, b2_c, (short)0, acc2, false, false);
    acc3 = __builtin_amdgcn_wmma_f32_16x16x4_f32(false, a_c, false, b3_c, (short)0, acc3, false, false);

    {
      long col;
      float bb;
      col = n0 + l;      bb = bias ? bias[col] : 0.f;
      #pragma unroll
      for (int r = 0; r < 8; ++r) Cp[(long)(m0 + r + 8 * hi) * ldc + col] = acc0[r] + bb;
      col = n0 + 16 + l; bb = bias ? bias[col] : 0.f;
      #pragma unroll
      for (int r = 0; r < 8; ++r) Cp[(long)(m0 + r + 8 * hi) * ldc + col] = acc1[r] + bb;
      col = n0 + 32 + l; bb = bias ? bias[col] : 0.f;
      #pragma unroll
      for (int r = 0; r < 8; ++r) Cp[(long)(m0 + r + 8 * hi) * ldc + col] = acc2[r] + bb;
      col = n0 + 48 + l; bb = bias ? bias[col] : 0.f;
      #pragma unroll
      for (int r = 0; r < 8; ++r) Cp[(long)(m0 + r + 8 * hi) * ldc + col] = acc3[r] + bb;
    }
  } else {
    // clamped-address / value-select path: unconditional loads, no exec churn
    int  rowc = row < Mm ? row : (Mm - 1);
    bool rowv = row < Mm;
    const float* arow = Ap + (long)rowc * lda;
    long cols[4], colc[4];
    bool cv[4];
    #pragma unroll
    for (int t = 0; t < 4; ++t) {
      cols[t] = n0 + 16 * t + l;
      cv[t]   = cols[t] < Nn;
      colc[t] = cv[t] ? cols[t] : (long)(Nn - 1);
    }
    for (int k = 0; k < Kk; k += 4) {
      int ka  = k + 2 * hi;
      int ka0 = ka     < Kk ? ka     : (Kk - 1);
      int ka1 = ka + 1 < Kk ? ka + 1 : (Kk - 1);
      bool kv0 = ka     < Kk;
      bool kv1 = ka + 1 < Kk;
      v2f a;
      float a0 = arow[ka0], a1 = arow[ka1];
      a.x = (rowv && kv0) ? a0 : 0.f;
      a.y = (rowv && kv1) ? a1 : 0.f;
      v2f bf[4];
      #pragma unroll
      for (int t = 0; t < 4; ++t) {
        float b0, b1;
        if (TB == 0) {
          b0 = Bp[(long)ka0 * ldb + colc[t]];
          b1 = Bp[(long)ka1 * ldb + colc[t]];
        } else {
          b0 = Bp[colc[t] * ldb + ka0];
          b1 = Bp[colc[t] * ldb + ka1];
        }
        bf[t].x = (cv[t] && kv0) ? b0 : 0.f;
        bf[t].y = (cv[t] && kv1) ? b1 : 0.f;
      }
      acc0 = __builtin_amdgcn_wmma_f32_16x16x4_f32(false, a, false, bf[0], (short)0, acc0, false, false);
      acc1 = __builtin_amdgcn_wmma_f32_16x16x4_f32(false, a, false, bf[1], (short)0, acc1, false, false);
      acc2 = __builtin_amdgcn_wmma_f32_16x16x4_f32(false, a, false, bf[2], (short)0, acc2, false, false);
      acc3 = __builtin_amdgcn_wmma_f32_16x16x4_f32(false, a, false, bf[3], (short)0, acc3, false, false);
    }
    #pragma unroll
    for (int t = 0; t < 4; ++t) {
      if (cols[t] < Nn) {
        float bb = bias ? bias[cols[t]] : 0.f;
        v8f av = (t == 0) ? acc0 : (t == 1) ? acc1 : (t == 2) ? acc2 : acc3;
        #pragma unroll
        for (int r = 0; r < 8; ++r) {
          int orow = m0 + r + 8 * hi;
          if (orow < Mm) Cp[(long)orow * ldc + cols[t]] = av[r] + bb;
        }
      }
    }
  }
}

// ---------------------------------------------------------------------------
// Train-mode BatchNorm over leading L rows, per channel c (grid = C blocks),
// followed by activation: 0=none, 1=lrelu(0.2), 2=relu.  In place.
// ---------------------------------------------------------------------------
__global__ void k_bn_act(float* __restrict__ z, long L, int C, int act) {
  int c = blockIdx.x;
  __shared__ float ssum[256], ssq[256];
  float s = 0.f, q = 0.f;
  for (long i = threadIdx.x; i < L; i += 256) {
    float v = z[i * C + c];
    s += v; q += v * v;
  }
  ssum[threadIdx.x] = s; ssq[threadIdx.x] = q;
  __syncthreads();
  for (int st = 128; st > 0; st >>= 1) {
    if ((int)threadIdx.x < st) {
      ssum[threadIdx.x] += ssum[threadIdx.x + st];
      ssq[threadIdx.x]  += ssq[threadIdx.x + st];
    }
    __syncthreads();
  }
  float mu  = ssum[0] / (float)L;
  float var = fmaxf(ssq[0] / (float)L - mu * mu, 0.f);
  float rs  = rsqrtf(var + 1e-5f);
  for (long i = threadIdx.x; i < L; i += 256) {
    float v = (z[i * C + c] - mu) * rs;
    if (act == 1)      v = v >= 0.f ? v : 0.2f * v;
    else if (act == 2) v = fmaxf(v, 0.f);
    z[i * C + c] = v;
  }
}

// x (B,3,N) -> xt (B,N,3)
__global__ void k_xt(const float* __restrict__ x, float* __restrict__ xt) {
  int i = blockIdx.x * 256 + threadIdx.x;
  if (i >= BB * 3 * NN) return;
  int n = i % NN; int t = i / NN; int c = t % 3; int b = t / 3;
  xt[((long)b * NN + n) * 3 + c] = x[i];
}

// cnt[b,p] = max(#points with p2v==pri[b,p], 1)
__global__ void k_cnt(const int* __restrict__ p2v, const int* __restrict__ pri,
                      float* __restrict__ cnt) {
  int bp = blockIdx.x; int b = bp / PP;
  int tgt = pri[bp];
  __shared__ int sc[256];
  int c = 0;
  for (int n = threadIdx.x; n < NN; n += 256) c += (p2v[b * NN + n] == tgt);
  sc[threadIdx.x] = c;
  __syncthreads();
  for (int st = 128; st > 0; st >>= 1) {
    if ((int)threadIdx.x < st) sc[threadIdx.x] += sc[threadIdx.x + st];
    __syncthreads();
  }
  if (threadIdx.x == 0) cnt[bp] = fmaxf((float)sc[0], 1.f);
}

// masked mean-pool: out[b,p,c] = sum_n mask * f[b,n,c] / cnt[b,p]
__global__ void k_pool(const float* __restrict__ f, int C,
                       const int* __restrict__ p2v, const int* __restrict__ pri,
                       const float* __restrict__ cnt, float* __restrict__ out) {
  int bp = blockIdx.x; int b = bp / PP;
  int tgt = pri[bp];
  int c = threadIdx.x;
  if (c >= C) return;
  float acc = 0.f;
  for (int n = 0; n < NN; ++n)
    if (p2v[b * NN + n] == tgt) acc += f[((long)b * NN + n) * C + c];
  out[(long)bp * C + c] = acc / cnt[bp];
}

// KNN: per (b,i) block, d2 row in LDS, 30 rounds of (d2, idx)-lexicographic argmin
__global__ void k_knn(const float* __restrict__ xt, int* __restrict__ knn) {
  int bi = blockIdx.x; int b = bi / NN;
  __shared__ float d2[NN];
  __shared__ float bv[256];
  __shared__ int   bj[256];
  float xi0 = xt[(long)bi * 3 + 0], xi1 = xt[(long)bi * 3 + 1], xi2 = xt[(long)bi * 3 + 2];
  float sqi = xi0 * xi0 + xi1 * xi1 + xi2 * xi2;
  for (int j = threadIdx.x; j < NN; j += 256) {
    long p = ((long)b * NN + j) * 3;
    float y0 = xt[p], y1 = xt[p + 1], y2 = xt[p + 2];
    float sqj = y0 * y0 + y1 * y1 + y2 * y2;
    d2[j] = sqi + sqj - 2.f * (xi0 * y0 + xi1 * y1 + xi2 * y2);
  }
  __syncthreads();
  for (int k = 0; k < KK; ++k) {
    float best = FLT_MAX; int bidx = NN;
    for (int j = threadIdx.x; j < NN; j += 256) {
      float v = d2[j];
      if (v < best || (v == best && j < bidx)) { best = v; bidx = j; }
    }
    bv[threadIdx.x] = best; bj[threadIdx.x] = bidx;
    __syncthreads();
    for (int st = 128; st > 0; st >>= 1) {
      if ((int)threadIdx.x < st) {
        float ov = bv[threadIdx.x + st]; int oj = bj[threadIdx.x + st];
        if (ov < bv[threadIdx.x] || (ov == bv[threadIdx.x] && oj < bj[threadIdx.x])) {
          bv[threadIdx.x] = ov; bj[threadIdx.x] = oj;
        }
      }
      __syncthreads();
    }
    if (threadIdx.x == 0) { knn[(long)bi * KK + k] = bj[0]; d2[bj[0]] = FLT_MAX; }
    __syncthreads();
  }
}

// build xyz (B,N,K,10) and gf (B,N,K,6)
__global__ void k_xyz(const float* __restrict__ xt, const int* __restrict__ knn,
                      float* __restrict__ xyz, float* __restrict__ gf) {
  int bn = blockIdx.x; int k = threadIdx.x;
  if (k >= KK) return;
  int b = bn / NN;
  int j = knn[(long)bn * KK + k];
  float c0 = xt[(long)bn * 3], c1 = xt[(long)bn * 3 + 1], c2 = xt[(long)bn * 3 + 2];
  long pj = ((long)b * NN + j) * 3;
  float n0 = xt[pj], n1 = xt[pj + 1], n2 = xt[pj + 2];
  float d0 = n0 - c0, d1 = n1 - c1, d2v = n2 - c2;
  float dist = sqrtf(fmaxf(d0 * d0 + d1 * d1 + d2v * d2v, 1e-12f));
  long xb = ((long)bn * KK + k) * 10;
  xyz[xb + 0] = c0; xyz[xb + 1] = c1; xyz[xb + 2] = c2;
  xyz[xb + 3] = n0; xyz[xb + 4] = n1; xyz[xb + 5] = n2;
  xyz[xb + 6] = d0; xyz[xb + 7] = d1; xyz[xb + 8] = d2v;
  xyz[xb + 9] = dist;
  long gb = ((long)bn * KK + k) * 6;
  gf[gb + 0] = d0; gf[gb + 1] = d1; gf[gb + 2] = d2v;
  gf[gb + 3] = c0; gf[gb + 4] = c1; gf[gb + 5] = c2;
}

// feat[b,n,o] = max_k e[(bn*K+k), o]
__global__ void k_maxk(const float* __restrict__ e, float* __restrict__ feat) {
  int bn = blockIdx.x; int o = threadIdx.x;
  float m = -FLT_MAX;
  for (int k = 0; k < KK; ++k) m = fmaxf(m, e[((long)bn * KK + k) * CH + o]);
  feat[(long)bn * CH + o] = m;
}

// Ksum = kernel[:64] + kernel[64:]
__global__ void k_ksum(const float* __restrict__ kern, float* __restrict__ ks) {
  int i = blockIdx.x * 256 + threadIdx.x;
  if (i >= CH * MM * CH) return;
  ks[i] = kern[i] + kern[CH * MM * CH + i];
}

// softmax over last dim of width 8
__global__ void k_softmax8(float* __restrict__ s, long rows) {
  long r = (long)blockIdx.x * 256 + threadIdx.x;
  if (r >= rows) return;
  float* p = s + r * MM;
  float v[MM]; float mx = -FLT_MAX;
  #pragma unroll
  for (int i = 0; i < MM; ++i) { v[i] = p[i]; mx = fmaxf(mx, v[i]); }
  float sum = 0.f;
  #pragma unroll
  for (int i = 0; i < MM; ++i) { v[i] = expf(v[i] - mx); sum += v[i]; }
  float inv = 1.f / sum;
  #pragma unroll
  for (int i = 0; i < MM; ++i) p[i] = v[i] * inv;
}

// out[b,n,o] = sum_{k,m} score[bn,k,m] * pt[b,idx,m,o]  -  sum_m(cen[bn,m,o]*sum_k score)
__global__ void k_paconv_out(const float* __restrict__ score, const float* __restrict__ pt,
                             const float* __restrict__ cen, const int* __restrict__ knn,
                             float* __restrict__ out) {
  int bn = blockIdx.x; int b = bn / NN; int o = threadIdx.x;
  __shared__ float s[KK * MM];
  __shared__ int   js[KK];
  for (int t = threadIdx.x; t < KK * MM; t += CH) s[t] = score[(long)bn * KK * MM + t];
  for (int t = threadIdx.x; t < KK; t += CH)      js[t] = knn[(long)bn * KK + t];
  __syncthreads();
  float acc = 0.f;
  float ssum[MM];
  #pragma unroll
  for (int m = 0; m < MM; ++m) ssum[m] = 0.f;
  for (int k = 0; k < KK; ++k) {
    const float* prow = pt + ((long)b * NN + js[k]) * (MM * CH) + o;
    #pragma unroll
    for (int m = 0; m < MM; ++m) {
      float sc = s[k * MM + m];
      acc += sc * prow[m * CH];
      ssum[m] += sc;
    }
  }
  const float* crow = cen + (long)bn * (MM * CH) + o;
  #pragma unroll
  for (int m = 0; m < MM; ++m) acc -= ssum[m] * crow[m * CH];
  out[(long)bn * CH + o] = acc;
}

// elementwise add
__global__ void k_add(const float* __restrict__ a, const float* __restrict__ b,
                      float* __restrict__ o, int n) {
  int i = blockIdx.x * 256 + threadIdx.x;
  if (i < n) o[i] = a[i] + b[i];
}

// e0[b,i,j,:] = [pf[b,j]-pf[b,i], pf[b,i]]
__global__ void k_e0(const float* __restrict__ pf, float* __restrict__ e0) {
  int bij = blockIdx.x; int c = threadIdx.x;       // 128 threads
  int b = bij / (PP * PP); int ij = bij % (PP * PP);
  int i = ij / PP, j = ij % PP;
  const float* pi = pf + ((long)b * PP + i) * CH;
  const float* pj = pf + ((long)b * PP + j) * CH;
  float v = (c < CH) ? (pj[c] - pi[c]) : pi[c - CH];
  e0[(long)bij * 128 + c] = v;
}

// per (b,i): t[j,h]=sum_d lrelu(g*e[...,4d+h])*w[d]; a=softmax_j(t); e *= a[j,h]
__global__ void k_att_scale(float* __restrict__ e, const float* __restrict__ attw,
                            const float* __restrict__ gh) {
  int bi = blockIdx.x;
  __shared__ float t[PP][4];
  __shared__ float a[PP][4];
  int tid = threadIdx.x;                            // 128 threads
  if (tid < PP * 4) {
    int j = tid >> 2, h = tid & 3;
    float g = gh ? gh[(long)bi * PP + j] : 1.f;
    const float* ep = e + ((long)bi * PP + j) * CH;
    float s = 0.f;
    #pragma unroll
    for (int d = 0; d < 16; ++d) {
      float v = g * ep[d * 4 + h];
      v = v >= 0.f ? v : 0.2f * v;
      s += v * attw[d];
    }
    t[j][h] = s;
  }
  __syncthreads();
  if (tid < 4) {
    int h = tid;
    float mx = -FLT_MAX;
    for (int j = 0; j < PP; ++j) mx = fmaxf(mx, t[j][h]);
    float sum = 0.f;
    for (int j = 0; j < PP; ++j) { float ev = expf(t[j][h] - mx); a[j][h] = ev; sum += ev; }
    float inv = 1.f / sum;
    for (int j = 0; j < PP; ++j) a[j][h] *= inv;
  }
  __syncthreads();
  for (int idx = tid; idx < PP * CH; idx += 128) {
    int j = idx >> 6, c = idx & 63;
    e[((long)bi * PP + j) * CH + c] *= a[j][c & 3];
  }
}

// hop = argmax(logits,5) (first max); gh = gauss(hop)
__global__ void k_hopgauss(const float* __restrict__ lg, float* __restrict__ gh, int n) {
  int i = blockIdx.x * 256 + threadIdx.x;
  if (i >= n) return;
  const float* p = lg + (long)i * 5;
  int am = 0; float bv = p[0];
  #pragma unroll
  for (int j = 1; j < 5; ++j) if (p[j] > bv) { bv = p[j]; am = j; }
  float h = (float)am;
  gh[i] = expf(-h * h * 0.5f) * 0.3989422804014327f;
}

// g[b,i,c] = mean_j e2[b,i,j,c]
__global__ void k_meanj(const float* __restrict__ e2, float* __restrict__ g) {
  int bi = blockIdx.x; int c = threadIdx.x;
  float s = 0.f;
  for (int j = 0; j < PP; ++j) s += e2[((long)bi * PP + j) * CH + c];
  g[(long)bi * CH + c] = s * (1.f / (float)PP);
}

// out[b,n,o] = y[b,n,o] + unpool(g)
__global__ void k_unpool_add(const float* __restrict__ y, const float* __restrict__ g,
                             const int* __restrict__ p2v, const int* __restrict__ pri,
                             float* __restrict__ out) {
  long i = (long)blockIdx.x * 256 + threadIdx.x;
  if (i >= (long)BB * NN * CH) return;
  int o = (int)(i & 63); long bn = i >> 6; int b = (int)(bn / NN);
  int pv = p2v[bn];
  float acc = 0.f;
  for (int p = 0; p < PP; ++p)
    if (pri[b * PP + p] == pv) acc += g[((long)b * PP + p) * CH + o];
  out[bn * CH + o] = y[bn * CH + o] + acc;
}

// concat x2..x5 along channels and transpose to (B,256,N)
__global__ void k_concat_out(const float* __restrict__ x2, const float* __restrict__ x3,
                             const float* __restrict__ x4, const float* __restrict__ x5,
                             float* __restrict__ out) {
  int bn = blockIdx.x; int c = threadIdx.x;        // 256 threads
  int b = bn / NN, n = bn % NN;
  const float* src = (c < 64) ? x2 : (c < 128) ? x3 : (c < 192) ? x4 : x5;
  out[((long)b * 256 + c) * NN + n] = src[(long)bn * CH + (c & 63)];
}

// ---------------------------------------------------------------------------
static void gemm(hipStream_t st, const float* A, int lda, const float* Bp, int ldb,
                 int tB, const float* bias, float* C, int ldc, int Mm, int Nn, int Kk) {
  dim3 g((Mm + 15) / 16, (Nn + 63) / 64);
  if (tB)
    k_gemm_wmma<1><<<g, 32, 0, st>>>(A, lda, Bp, ldb, bias, C, ldc, Mm, Nn, Kk);
  else
    k_gemm_wmma<0><<<g, 32, 0, st>>>(A, lda, Bp, ldb, bias, C, ldc, Mm, Nn, Kk);
}

extern "C" void kernel_launch(void* const* d_in, const int* in_sizes, int n_in,
                              void* d_out, int out_size, void* d_ws, size_t ws_size,
                              hipStream_t stream) {
  if (n_in < 46) return;
  const float *x, *conv_pce_w, *conv1_w, *conv1_b;
  const float *kern[4], *snw1[4], *snw2[4], *snb2[4];
  const float *ec1[4], *hw1[4], *hw2[4], *ec2[4], *aw1[4], *aw2[4];
  const int *p2v, *pri;
  if (in_sizes[0] == BB * 3 * NN) {
    // insertion-order flattening: x, params{...}, p2v, pri
    x = (const float*)d_in[0];
    conv_pce_w = (const float*)d_in[1];
    conv1_w = (const float*)d_in[2];
    conv1_b = (const float*)d_in[3];
    for (int i = 0; i < 4; ++i) {
      kern[i] = (const float*)d_in[4 + 4 * i];
      snw1[i] = (const float*)d_in[5 + 4 * i];
      snw2[i] = (const float*)d_in[6 + 4 * i];
      snb2[i] = (const float*)d_in[7 + 4 * i];
    }
    for (int i = 0; i < 4; ++i) {
      ec1[i] = (const float*)d_in[20 + 6 * i];
      hw1[i] = (const float*)d_in[21 + 6 * i];
      hw2[i] = (const float*)d_in[22 + 6 * i];
      ec2[i] = (const float*)d_in[23 + 6 * i];
      aw1[i] = (const float*)d_in[24 + 6 * i];
      aw2[i] = (const float*)d_in[25 + 6 * i];
    }
    p2v = (const int*)d_in[44];
    pri = (const int*)d_in[45];
  } else {
    // sorted tree_leaves flattening
    p2v = (const int*)d_in[0];
    conv1_b = (const float*)d_in[1];
    conv1_w = (const float*)d_in[2];
    conv_pce_w = (const float*)d_in[3];
    for (int i = 0; i < 4; ++i) {
      aw1[i] = (const float*)d_in[4 + 6 * i];
      aw2[i] = (const float*)d_in[5 + 6 * i];
      ec1[i] = (const float*)d_in[6 + 6 * i];
      ec2[i] = (const float*)d_in[7 + 6 * i];
      hw1[i] = (const float*)d_in[8 + 6 * i];
      hw2[i] = (const float*)d_in[9 + 6 * i];
    }
    for (int i = 0; i < 4; ++i) kern[i] = (const float*)d_in[28 + i];
    for (int i = 0; i < 4; ++i) {
      snb2[i] = (const float*)d_in[32 + 3 * i];
      snw1[i] = (const float*)d_in[33 + 3 * i];
      snw2[i] = (const float*)d_in[34 + 3 * i];
    }
    pri = (const int*)d_in[44];
    x = (const float*)d_in[45];
  }

  // ---- workspace layout ----
  char* base = (char*)d_ws;
  size_t off = 0;
  auto alloc = [&](size_t bytes) -> void* {
    void* p = base + off;
    off = (off + bytes + 255) & ~(size_t)255;
    return p;
  };
  const long LNK = (long)BB * NN * KK;           // 245760
  float* xt    = (float*)alloc(sizeof(float) * BB * NN * 3);
  float* cnt   = (float*)alloc(sizeof(float) * BB * PP);
  float* poolx = (float*)alloc(sizeof(float) * BB * PP * 3);
  float* pce   = (float*)alloc(sizeof(float) * BB * PP * CH);
  int*   knn   = (int*)  alloc(sizeof(int)   * LNK);
  float* xyz   = (float*)alloc(sizeof(float) * LNK * 10);
  float* feat  = (float*)alloc(sizeof(float) * BB * NN * CH);
  float* xb[4];
  for (int i = 0; i < 4; ++i) xb[i] = (float*)alloc(sizeof(float) * BB * NN * CH);
  float* ybuf  = (float*)alloc(sizeof(float) * BB * NN * CH);
  float* pooly = (float*)alloc(sizeof(float) * BB * PP * CH);
  float* padd  = (float*)alloc(sizeof(float) * BB * PP * CH);
  float* gpart = (float*)alloc(sizeof(float) * BB * PP * CH);
  float* ksum  = (float*)alloc(sizeof(float) * CH * MM * CH);
  float* e0    = (float*)alloc(sizeof(float) * BB * PP * PP * 128);
  float* e1    = (float*)alloc(sizeof(float) * BB * PP * PP * CH);
  float* hh    = (float*)alloc(sizeof(float) * BB * PP * PP * 32);
  float* lgt   = (float*)alloc(sizeof(float) * BB * PP * PP * 5);
  float* gh    = (float*)alloc(sizeof(float) * BB * PP * PP);
  float* e2    = (float*)alloc(sizeof(float) * BB * PP * PP * CH);
  // stage-overlaid scratch (max of conv1 stage and dh stage)
  float* scratch = (float*)alloc(sizeof(float) * (size_t)(LNK * 6 + LNK * CH));
  float* gf     = scratch;                        // (B,N,K,6)
  float* econv1 = scratch + LNK * 6;              // (B,N,K,64)
  float* pt     = scratch;                        // (B,N,8,64)
  float* cen    = pt  + (long)BB * NN * MM * CH;  // (B,N,8,64)
  float* h1     = cen + (long)BB * NN * MM * CH;  // (B,N,K,16)
  float* score  = h1  + LNK * 16;                 // (B,N,K,8)
  (void)ws_size; (void)out_size; (void)in_sizes;

  const int NPART = BB * PP;                      // 216
  const int NEDGE = BB * PP * PP;                 // 5832

  // ---- pipeline ----
  k_xt<<<(BB * 3 * NN + 255) / 256, 256, 0, stream>>>(x, xt);
  k_cnt<<<NPART, 256, 0, stream>>>(p2v, pri, cnt);
  k_pool<<<NPART, 32, 0, stream>>>(xt, 3, p2v, pri, cnt, poolx);
  gemm(stream, poolx, 3, conv_pce_w, 3, 1, nullptr, pce, CH, NPART, CH, 3);
  k_bn_act<<<CH, 256, 0, stream>>>(pce, NPART, CH, 1);

  k_knn<<<BB * NN, 256, 0, stream>>>(xt, knn);
  k_xyz<<<BB * NN, 32, 0, stream>>>(xt, knn, xyz, gf);

  gemm(stream, gf, 6, conv1_w, 6, 1, conv1_b, econv1, CH, (int)LNK, CH, 6);
  k_bn_act<<<CH, 256, 0, stream>>>(econv1, LNK, CH, 1);
  k_maxk<<<BB * NN, CH, 0, stream>>>(econv1, feat);

  const float* cur = feat;
  for (int blk = 0; blk < 4; ++blk) {
    // pt = f @ (K0+K1), cen = f @ K0  (WMMA GEMMs)
    k_ksum<<<(CH * MM * CH + 255) / 256, 256, 0, stream>>>(kern[blk], ksum);
    gemm(stream, cur, CH, ksum,      MM * CH, 0, nullptr, pt,  MM * CH, BB * NN, MM * CH, CH);
    gemm(stream, cur, CH, kern[blk], MM * CH, 0, nullptr, cen, MM * CH, BB * NN, MM * CH, CH);
    // scorenet
    gemm(stream, xyz, 10, snw1[blk], 10, 1, nullptr, h1, 16, (int)LNK, 16, 10);
    k_bn_act<<<16, 256, 0, stream>>>(h1, LNK, 16, 2);
    gemm(stream, h1, 16, snw2[blk], 16, 1, snb2[blk], score, MM, (int)LNK, MM, 16);
    k_softmax8<<<(int)((LNK + 255) / 256), 256, 0, stream>>>(score, LNK);
    // gather-einsum + BN(lrelu)
    k_paconv_out<<<BB * NN, CH, 0, stream>>>(score, pt, cen, knn, ybuf);
    k_bn_act<<<CH, 256, 0, stream>>>(ybuf, (long)BB * NN, CH, 1);
    // part pooling + hopgcn
    k_pool<<<NPART, CH, 0, stream>>>(ybuf, CH, p2v, pri, cnt, pooly);
    k_add<<<(NPART * CH + 255) / 256, 256, 0, stream>>>(pce, pooly, padd, NPART * CH);
    k_e0<<<NEDGE, 128, 0, stream>>>(padd, e0);
    gemm(stream, e0, 128, ec1[blk], 128, 1, nullptr, e1, CH, NEDGE, CH, 128);
    k_bn_act<<<CH, 256, 0, stream>>>(e1, NEDGE, CH, 1);
    k_att_scale<<<NPART, 128, 0, stream>>>(e1, aw1[blk], nullptr);
    gemm(stream, e1, CH, hw1[blk], CH, 1, nullptr, hh, 32, NEDGE, 32, CH);
    k_bn_act<<<32, 256, 0, stream>>>(hh, NEDGE, 32, 1);
    gemm(stream, hh, 32, hw2[blk], 32, 1, nullptr, lgt, 5, NEDGE, 5, 32);
    k_bn_act<<<5, 256, 0, stream>>>(lgt, NEDGE, 5, 1);
    k_hopgauss<<<(NEDGE + 255) / 256, 256, 0, stream>>>(lgt, gh, NEDGE);
    gemm(stream, e1, CH, ec2[blk], CH, 1, nullptr, e2, CH, NEDGE, CH, CH);
    k_bn_act<<<CH, 256, 0, stream>>>(e2, NEDGE, CH, 1);
    k_att_scale<<<NPART, 128, 0, stream>>>(e2, aw2[blk], gh);
    k_meanj<<<NPART, CH, 0, stream>>>(e2, gpart);
    // residual unpool
    k_unpool_add<<<(int)(((long)BB * NN * CH + 255) / 256), 256, 0, stream>>>(
        ybuf, gpart, p2v, pri, xb[blk]);
    cur = xb[blk];
  }

  k_concat_out<<<BB * NN, 256, 0, stream>>>(xb[0], xb[1], xb[2], xb[3], (float*)d_out);
}